// Model_15436112462638
// MI455X (gfx1250) — compile-verified
//
#include <hip/hip_runtime.h>
#include <hip/hip_bf16.h>

typedef __attribute__((ext_vector_type(2))) float v2f;
typedef __attribute__((ext_vector_type(8))) float v8f;

#define F 128           // feature dim (in == out)
#define M_HEDGES 50000  // NUM_HYPEREDGES (fixed by reference)

// ---------------------------------------------------------------------------
// Zero scratch accumulators (m, deg_v, deg_e) and d_out (harness poisons them)
// ---------------------------------------------------------------------------
__global__ void zero_kernel(float* m, float* out, float* dv, float* de,
                            int nf, int n_nodes, int n_hedges) {
    int i = blockIdx.x * blockDim.x + threadIdx.x;
    if (i < nf) { m[i] = 0.f; out[i] = 0.f; }
    if (i < n_nodes)  dv[i] = 0.f;
    if (i < n_hedges) de[i] = 0.f;
}

// ---------------------------------------------------------------------------
// Degree counts: one thread per incidence edge
// ---------------------------------------------------------------------------
__global__ void degree_kernel(const int* __restrict__ nodes,
                              const int* __restrict__ hedges,
                              float* dv, float* de, int E) {
    int i = blockIdx.x * blockDim.x + threadIdx.x;
    if (i >= E) return;
    unsafeAtomicAdd(&dv[nodes[i]], 1.0f);
    unsafeAtomicAdd(&de[hedges[i]], 1.0f);
}

// ---------------------------------------------------------------------------
// x = X @ W via V_WMMA_F32_16X16X4_F32.
// Block = 256 threads = 8 waves. Block owns a 16-row tile of X; wave w owns
// output columns [16w,16w+16). X tile (16x128, 8KB) and W K-chunks (32x128,
// 16KB) staged in LDS. K loop: 4 chunks x 8 sub-steps of K=4 -> 32 WMMAs/wave.
// 50000 rows = 3125 * 16 exactly (no remainder handling needed).
// ---------------------------------------------------------------------------
__global__ __launch_bounds__(256) void gemm_wmma_f32(
    const float* __restrict__ X, const float* __restrict__ W,
    float* __restrict__ Y) {
    __shared__ float Xl[16 * F];   // 8 KB
    __shared__ float Wl[32 * F];   // 16 KB

    const int tid  = threadIdx.x;
    const int wv   = tid >> 5;         // wave id 0..7 -> column tile
    const int lane = tid & 31;
    const int half = lane >> 4;        // 0: K even pair, 1: K odd pair
    const int l15  = lane & 15;
    const int r0   = blockIdx.x * 16;  // row tile base
    const int ncol = wv * 16 + l15;    // this lane's output column

    // Cooperative load of the 16x128 X tile (512 float4, 2 per thread)
    {
        const float4* src = (const float4*)(X + (size_t)r0 * F);
        float4* dst = (float4*)Xl;
        dst[tid]       = src[tid];
        dst[tid + 256] = src[tid + 256];
    }

    v8f c = {};  // 16x16 f32 accumulator (8 VGPRs)

    for (int kc = 0; kc < 4; ++kc) {
        __syncthreads();  // Xl ready (kc==0) / Wl consumed (kc>0)
        // Load W rows [kc*32, kc*32+32) : 1024 float4, 4 per thread
        const float4* wsrc = (const float4*)(W + (size_t)kc * 32 * F);
        float4* wdst = (float4*)Wl;
#pragma unroll
        for (int i = 0; i < 4; ++i) wdst[tid + 256 * i] = wsrc[tid + 256 * i];
        __syncthreads();

#pragma unroll
        for (int kk = 0; kk < 8; ++kk) {
            const int k  = kc * 32 + kk * 4;   // global K base of this step
            const int kb = kk * 4 + 2 * half;  // K index within LDS chunk
            // A 16x4 fragment: lanes 0-15 hold M=l15,K={0,1}; lanes 16-31 K={2,3}
            v2f a = *(const v2f*)&Xl[l15 * F + k + 2 * half];
            // B 4x16 fragment: symmetric layout, one column per lane
            v2f b;
            b.x = Wl[kb * F + ncol];
            b.y = Wl[(kb + 1) * F + ncol];
            c = __builtin_amdgcn_wmma_f32_16x16x4_f32(
                    false, a, false, b, (short)0, c, false, false);
        }
    }

    // D layout: lanes 0-15 -> M=r, N=lane; lanes 16-31 -> M=r+8, N=lane-16
#pragma unroll
    for (int r = 0; r < 8; ++r) {
        const int mrow = r + 8 * half;
        Y[(size_t)(r0 + mrow) * F + ncol] = c[r];
    }
}

// ---------------------------------------------------------------------------
// Scatter 1: m[hedge] += x[node].  One wave32 per edge; each lane owns a
// float4 (4 features) -> fully coalesced 128-float row per wave.
// ---------------------------------------------------------------------------
__global__ void scatter_node_to_hedge(const int* __restrict__ nodes,
                                      const int* __restrict__ hedges,
                                      const float* __restrict__ x,
                                      float* __restrict__ m, int E) {
    const int gid  = blockIdx.x * blockDim.x + threadIdx.x;
    const int e    = gid >> 5;
    const int lane = gid & 31;
    if (e >= E) return;
    const int n = nodes[e];
    const int h = hedges[e];
    const float4 v = *(const float4*)(x + (size_t)n * F + lane * 4);
    float* dst = m + (size_t)h * F + lane * 4;
    unsafeAtomicAdd(dst + 0, v.x);
    unsafeAtomicAdd(dst + 1, v.y);
    unsafeAtomicAdd(dst + 2, v.z);
    unsafeAtomicAdd(dst + 3, v.w);
}

// ---------------------------------------------------------------------------
// Scatter 2: out[node] += m[hedge] * (1/deg_e[hedge]).  B^-1 fused into the
// gather so m never needs a separate scaling pass. deg_e[h] >= 1 whenever h
// appears in the edge list, so no zero-guard needed here.
// ---------------------------------------------------------------------------
__global__ void scatter_hedge_to_node(const int* __restrict__ nodes,
                                      const int* __restrict__ hedges,
                                      const float* __restrict__ m,
                                      const float* __restrict__ de,
                                      float* __restrict__ out, int E) {
    const int gid  = blockIdx.x * blockDim.x + threadIdx.x;
    const int e    = gid >> 5;
    const int lane = gid & 31;
    if (e >= E) return;
    const int n = nodes[e];
    const int h = hedges[e];
    const float binv = 1.0f / de[h];
    const float4 v = *(const float4*)(m + (size_t)h * F + lane * 4);
    float* dst = out + (size_t)n * F + lane * 4;
    unsafeAtomicAdd(dst + 0, v.x * binv);
    unsafeAtomicAdd(dst + 1, v.y * binv);
    unsafeAtomicAdd(dst + 2, v.z * binv);
    unsafeAtomicAdd(dst + 3, v.w * binv);
}

// ---------------------------------------------------------------------------
// Finalize: out = softmax(out * D^-1 + b) per row. One wave32 per row,
// shfl_xor reductions across the 32 lanes (4 features per lane).
// ---------------------------------------------------------------------------
__global__ void finalize_softmax(float* __restrict__ out,
                                 const float* __restrict__ dv,
                                 const float* __restrict__ b, int N) {
    const int gid  = blockIdx.x * blockDim.x + threadIdx.x;
    const int row  = gid >> 5;
    const int lane = gid & 31;
    if (row >= N) return;

    const float d    = dv[row];
    const float dinv = d > 0.f ? 1.0f / d : 0.0f;

    float4 acc      = *(float4*)(out + (size_t)row * F + lane * 4);
    const float4 bb = *(const float4*)(b + lane * 4);
    float v0 = acc.x * dinv + bb.x;
    float v1 = acc.y * dinv + bb.y;
    float v2 = acc.z * dinv + bb.z;
    float v3 = acc.w * dinv + bb.w;

    float mx = fmaxf(fmaxf(v0, v1), fmaxf(v2, v3));
#pragma unroll
    for (int off = 16; off > 0; off >>= 1)
        mx = fmaxf(mx, __shfl_xor(mx, off, 32));

    v0 = expf(v0 - mx); v1 = expf(v1 - mx);
    v2 = expf(v2 - mx); v3 = expf(v3 - mx);
    float s = v0 + v1 + v2 + v3;
#pragma unroll
    for (int off = 16; off > 0; off >>= 1)
        s += __shfl_xor(s, off, 32);

    const float rinv = 1.0f / s;
    acc.x = v0 * rinv; acc.y = v1 * rinv;
    acc.z = v2 * rinv; acc.w = v3 * rinv;
    *(float4*)(out + (size_t)row * F + lane * 4) = acc;
}

// ---------------------------------------------------------------------------
// Launch
// ---------------------------------------------------------------------------
extern "C" void kernel_launch(void* const* d_in, const int* in_sizes, int n_in,
                              void* d_out, int out_size, void* d_ws, size_t ws_size,
                              hipStream_t stream) {
    const float* X  = (const float*)d_in[0];
    const int*   ei = (const int*)d_in[1];      // [2, E] stacked: nodes then hedges
    const float* W  = (const float*)d_in[2];
    const float* b  = (const float*)d_in[3];

    const int N  = in_sizes[0] / F;   // 50000 nodes
    const int E  = in_sizes[1] / 2;   // 1,600,000 incidences
    const int Mh = M_HEDGES;          // 50000 hyperedges
    const int NF = N * F;

    const int* nodes  = ei;
    const int* hedges = ei + E;

    float* ws    = (float*)d_ws;
    float* x_lin = ws;                 // [N, F]
    float* m     = ws + (size_t)NF;    // [Mh, F]
    float* dv    = m + (size_t)Mh * F; // [N]
    float* de    = dv + N;             // [Mh]
    float* out   = (float*)d_out;      // [N, F] accumulate then softmax in place

    // 1) zero accumulators
    zero_kernel<<<(NF + 255) / 256, 256, 0, stream>>>(m, out, dv, de, NF, N, Mh);
    // 2) degrees
    degree_kernel<<<(E + 255) / 256, 256, 0, stream>>>(nodes, hedges, dv, de, E);
    // 3) x = X @ W  (WMMA fp32)
    gemm_wmma_f32<<<N / 16, 256, 0, stream>>>(X, W, x_lin);
    // 4) node -> hyperedge
    {
        long long th = (long long)E * 32;
        scatter_node_to_hedge<<<(unsigned)((th + 255) / 256), 256, 0, stream>>>(
            nodes, hedges, x_lin, m, E);
    }
    // 5) hyperedge -> node (B^-1 fused)
    {
        long long th = (long long)E * 32;
        scatter_hedge_to_node<<<(unsigned)((th + 255) / 256), 256, 0, stream>>>(
            nodes, hedges, m, de, out, E);
    }
    // 6) out = softmax(out * D^-1 + b)
    {
        long long th = (long long)N * 32;
        finalize_softmax<<<(unsigned)((th + 255) / 256), 256, 0, stream>>>(
            out, dv, b, N);
    }
}